// ProxyNCALayer_43585328120110
// MI455X (gfx1250) — compile-verified
//
#include <hip/hip_runtime.h>
#include <hip/hip_bf16.h>
#include <math.h>

// ProxyNCA loss, fused streaming-logsumexp f32 GEMM on gfx1250 WMMA.
//
// loss_i = -s[i,t_i] + logsumexp_{k != t_i} s[i,k]
//   where s[i,k] = scl[k]*dot(xs_i, proxies_k) - psq[k]
//   scl[k] = 2/max(||p_k||,1e-12), psq[k] = sum(p_norm_k^2)
// (x_sq cancels between the pos term and the logsumexp shift.)
//
// Pass 1 precomputes {scl, psq} per proxy. Pass 2 runs the WMMA GEMM with a
// branchless per-lane online logsumexp over ALL columns; the target column's
// contribution is computed directly in the epilogue (one 64-dot per row) and
// subtracted from the denominator: S_excl = S - exp(s_t - M). Pass 3 is a
// deterministic 1024 -> 1 mean reduction.

typedef __attribute__((ext_vector_type(2))) float v2f;
typedef __attribute__((ext_vector_type(4))) float v4f;
typedef __attribute__((ext_vector_type(8))) float v8f;

#define B_ROWS   1024
#define C_TOTAL  100000
#define D_DIM    64
#define NTILES   (C_TOTAL / 16)   // 6250
#define WAVES    16               // 512 threads / wave32

// ---------------------------------------------------------------- pass 1
// Per-proxy normalization constants: norm2[k] = { 2/max(||p||,eps), psq }.
__global__ __launch_bounds__(256)
void proxynca_norm(const float* __restrict__ proxies,
                   v2f*         __restrict__ norm2) {
  const int i = blockIdx.x * 256 + threadIdx.x;
  if (i >= C_TOTAL) return;
  const float* p = proxies + (size_t)i * D_DIM;
  float tot = 0.0f;
#pragma unroll
  for (int j = 0; j < 16; ++j) {
    const v4f q = *(const v4f*)(p + j * 4);
    tot += q.x * q.x + q.y * q.y + q.z * q.z + q.w * q.w;
  }
  const float inv = 1.0f / fmaxf(sqrtf(tot), 1e-12f);
  v2f r;
  r.x = 2.0f * inv;          // scl
  r.y = (tot * inv) * inv;   // psq (== 1 unless degenerate row)
  norm2[i] = r;
}

// ---------------------------------------------------------------- pass 2
__global__ __launch_bounds__(512, 1)
void proxynca_main(const float* __restrict__ xs,
                   const int*   __restrict__ ts,
                   const float* __restrict__ proxies,
                   const v2f*   __restrict__ norm2,
                   float*       __restrict__ row_loss) {
  __shared__ float red_m[16 * WAVES]; // per-row running max, per wave
  __shared__ float red_s[16 * WAVES]; // per-row running sum, per wave

  const int tid    = threadIdx.x;
  const int wave   = tid >> 5;
  const int lane   = tid & 31;
  const int half   = lane >> 4;   // which K-pair half (ISA A/B layout)
  const int lcol   = lane & 15;   // matrix row (A) / column (B, C/D)
  const int m_base = blockIdx.x * 16;

  // A fragments: A[m][k] = xs[m_base+m][k], m = lcol, k-pair = 4j + 2*half.
  // Held in registers for the entire N loop.
  v2f a[16];
  const float* xrow = xs + (size_t)(m_base + lcol) * D_DIM + half * 2;
#pragma unroll
  for (int j = 0; j < 16; ++j) a[j] = *(const v2f*)(xrow + j * 4);

  // Per-lane online-logsumexp state over this lane's columns, 8 rows each.
  float mx[8], sm[8];
#pragma unroll
  for (int v = 0; v < 8; ++v) { mx[v] = -__builtin_inff(); sm[v] = 0.0f; }

  for (int t = wave; t < NTILES; t += WAVES) {
    const int n = t * 16 + lcol;                    // proxy (column) index
    const float* prow = proxies + (size_t)n * D_DIM + half * 2;

    // Prefetch next tile for this wave into cache (global_prefetch_b8).
    if (t + WAVES < NTILES)
      __builtin_prefetch(prow + (size_t)WAVES * 16 * D_DIM, 0, 0);

    // B fragments: B[k][n] = proxies[n][k]; plus this column's {scl, psq}.
    const v2f sp = norm2[n];
    v2f b[16];
#pragma unroll
    for (int j = 0; j < 16; ++j) b[j] = *(const v2f*)(prow + j * 4);

    // 16 chained f32 WMMAs: full K=64 dot for the 16x16 tile.
    v8f acc = {};
#pragma unroll
    for (int j = 0; j < 16; ++j)
      acc = __builtin_amdgcn_wmma_f32_16x16x4_f32(
          /*neg_a=*/false, a[j], /*neg_b=*/false, b[j],
          /*c_mod=*/(short)0, acc, /*reuse_a=*/false, /*reuse_b=*/false);

    // Per-lane fixup: column n, rows m_base + v + 8*half.
    // Branchless single-exp online logsumexp update (no target test here).
#pragma unroll
    for (int v = 0; v < 8; ++v) {
      const float s  = acc[v] * sp.x - sp.y;
      const float d  = s - mx[v];
      const float e  = __expf(-fabsf(d));     // exp(-inf)=0 handles first iter
      const bool  gt = d > 0.0f;
      sm[v] = sm[v] * (gt ? e : 1.0f) + (gt ? 1.0f : e);
      mx[v] = fmaxf(mx[v], s);
    }
  }

  // Merge (max,sum) across the 16 lanes that share each row (xor 1,2,4,8
  // stays within a half; the two halves hold different rows).
#pragma unroll
  for (int v = 0; v < 8; ++v) {
#pragma unroll
    for (int off = 1; off < 16; off <<= 1) {
      const float mo = __shfl_xor(mx[v], off, 32);
      const float so = __shfl_xor(sm[v], off, 32);
      const float M  = fmaxf(mx[v], mo);
      sm[v] = sm[v] * __expf(mx[v] - M) + so * __expf(mo - M);
      mx[v] = M;
    }
  }
  if (lcol == 0) {  // lane 0 -> rows 0..7, lane 16 -> rows 8..15
#pragma unroll
    for (int v = 0; v < 8; ++v) {
      const int mrow = v + (half << 3);
      red_m[mrow * WAVES + wave] = mx[v];
      red_s[mrow * WAVES + wave] = sm[v];
    }
  }
  __syncthreads();

  // Merge across the 16 waves; compute the target score directly and remove
  // its contribution from the denominator; emit per-row loss.
  if (tid < 16) {
    float M = red_m[tid * WAVES];
    float S = red_s[tid * WAVES];
    for (int w = 1; w < WAVES; ++w) {
      const float mo = red_m[tid * WAVES + w];
      const float so = red_s[tid * WAVES + w];
      const float Mn = fmaxf(M, mo);
      S = S * __expf(M - Mn) + so * __expf(mo - Mn);
      M = Mn;
    }
    // s_t = scl[t]*dot(xs_row, proxies[t]) - psq[t]
    const int    tgt = ts[m_base + tid];
    const float* xr  = xs + (size_t)(m_base + tid) * D_DIM;
    const float* pr  = proxies + (size_t)tgt * D_DIM;
    float dot = 0.0f;
#pragma unroll
    for (int j = 0; j < 16; ++j) {
      const v4f qa = *(const v4f*)(xr + j * 4);
      const v4f qb = *(const v4f*)(pr + j * 4);
      dot += qa.x * qb.x + qa.y * qb.y + qa.z * qb.z + qa.w * qb.w;
    }
    const v2f  spt = norm2[tgt];
    const float st = dot * spt.x - spt.y;
    const float Se = fmaxf(S - __expf(st - M), 1e-30f);  // exclude target
    row_loss[m_base + tid] = -st + M + logf(Se);
  }
}

// ---------------------------------------------------------------- pass 3
// Deterministic 1024 -> 1 mean reduction (no float atomics).
__global__ __launch_bounds__(256)
void proxynca_reduce(const float* __restrict__ row_loss,
                     float*       __restrict__ out) {
  __shared__ float part[8];
  const int tid = threadIdx.x;
  float v = row_loss[tid] + row_loss[tid + 256] +
            row_loss[tid + 512] + row_loss[tid + 768];
#pragma unroll
  for (int off = 16; off > 0; off >>= 1) v += __shfl_xor(v, off, 32);
  if ((tid & 31) == 0) part[tid >> 5] = v;
  __syncthreads();
  if (tid < 8) {
    float x = part[tid];
    x += __shfl_xor(x, 1, 32);
    x += __shfl_xor(x, 2, 32);
    x += __shfl_xor(x, 4, 32);
    if (tid == 0) out[0] = x * (1.0f / (float)B_ROWS);
  }
}

extern "C" void kernel_launch(void* const* d_in, const int* in_sizes, int n_in,
                              void* d_out, int out_size, void* d_ws, size_t ws_size,
                              hipStream_t stream) {
  const float* xs      = (const float*)d_in[0];  // [1024, 64] f32
  const int*   ts      = (const int*)d_in[1];    // [1024] int
  const float* proxies = (const float*)d_in[2];  // [100000, 64] f32
  float*       out     = (float*)d_out;          // scalar f32
  // ws layout: [0, 4KB)  row_loss (1024 f32)
  //            [4KB, ...) norm2   (100000 v2f = 800KB)
  float* rloss = (float*)d_ws;
  v2f*   norm2 = (v2f*)((char*)d_ws + B_ROWS * sizeof(float));

  proxynca_norm<<<(C_TOTAL + 255) / 256, 256, 0, stream>>>(proxies, norm2);
  proxynca_main<<<B_ROWS / 16, 512, 0, stream>>>(xs, ts, proxies, norm2, rloss);
  proxynca_reduce<<<1, 256, 0, stream>>>(rloss, out);
}